// GIN_COO_22127671509526
// MI455X (gfx1250) — compile-verified
//
#include <hip/hip_runtime.h>
#include <hip/hip_bf16.h>

typedef __attribute__((ext_vector_type(2))) float v2f;
typedef __attribute__((ext_vector_type(8))) float v8f;

#define NN      100000
#define EE      1600000
#define FEAT    64
#define EDIM    16
#define NCLS    7

// ---------------------------------------------------------------------------
// utility kernels
// ---------------------------------------------------------------------------
__global__ void zero_kernel(float* __restrict__ p, long n) {
    long i = (long)blockIdx.x * blockDim.x + threadIdx.x;
    long stride = (long)gridDim.x * blockDim.x;
    for (; i < n; i += stride) p[i] = 0.0f;
}

__global__ void deg_kernel(const int* __restrict__ dst, float* __restrict__ deg, int e) {
    long i = (long)blockIdx.x * blockDim.x + threadIdx.x;
    long stride = (long)gridDim.x * blockDim.x;
    for (; i < e; i += stride) atomicAdd(&deg[dst[i]], 1.0f);
}

__global__ void dinv_kernel(float* __restrict__ deg, int n) {
    int i = blockIdx.x * blockDim.x + threadIdx.x;
    if (i < n) deg[i] = rsqrtf(deg[i] + 1.0f);   // +1 for self loop
}

// ---------------------------------------------------------------------------
// GINE fused edge kernel: msg = relu(x[src] + edge_attr@We + be); agg[dst]+=msg
// One wave handles a tile of 16 edges. WMMA f32 16x16x4, K=16 as 4 chunks,
// 64 output features as 4 N-tiles. W is stored in LDS with rows k/k+1
// interleaved so each B fragment is a single aligned ds_load_b64.
// ---------------------------------------------------------------------------
__global__ __launch_bounds__(256)
void gine_edge_kernel(const float* __restrict__ x,      // [NN,64]
                      const float* __restrict__ eattr,  // [EE,16]
                      const float* __restrict__ We,     // [16,64]
                      const float* __restrict__ be,     // [64]
                      const int*   __restrict__ src,
                      const int*   __restrict__ dst,
                      float*       __restrict__ agg,    // [NN,64]
                      int nTiles)
{
    // Wp[kp][n][2] : Wp[kp*128 + 2n + j] = We[2*kp + j][n]
    __shared__ float Wp[EDIM * FEAT];                   // 4 KB
    __shared__ float bes[FEAT];
    for (int i = threadIdx.x; i < (EDIM / 2) * FEAT; i += 256) {
        const int kp = i >> 6;                          // row pair
        const int n  = i & 63;
        Wp[kp * 128 + 2 * n + 0] = We[(2 * kp    ) * FEAT + n];
        Wp[kp * 128 + 2 * n + 1] = We[(2 * kp + 1) * FEAT + n];
    }
    if (threadIdx.x < FEAT) bes[threadIdx.x] = be[threadIdx.x];
    __syncthreads();

    int wave = (blockIdx.x * 256 + threadIdx.x) >> 5;
    int lane = threadIdx.x & 31;
    if (wave >= nTiles) return;                         // uniform per wave

    const int eBase = wave * 16;
    const int hi = lane >> 4;                           // lane half
    const int ln = lane & 15;

    v8f acc0 = {0,0,0,0,0,0,0,0};
    v8f acc[4] = {acc0, acc0, acc0, acc0};

    // A: edge_attr tile, row m = ln, K pair {kc, kc+1}, kc = 4c + 2*hi (even)
    const float* arow = &eattr[(size_t)(eBase + ln) * EDIM];
    #pragma unroll
    for (int c = 0; c < 4; ++c) {
        const int kc = 4 * c + 2 * hi;
        const v2f a = *(const v2f*)&arow[kc];
        #pragma unroll
        for (int t = 0; t < 4; ++t) {
            const v2f b = *(const v2f*)&Wp[(kc >> 1) * 128 + 2 * (16 * t + ln)];
            acc[t] = __builtin_amdgcn_wmma_f32_16x16x4_f32(
                false, a, false, b, (short)0, acc[t], false, false);
        }
    }

    // C layout: lane = N (ln), vgpr r = M (m = r + 8*hi)
    #pragma unroll
    for (int r = 0; r < 8; ++r) {
        const int m = r + 8 * hi;
        const int e = eBase + m;
        const int s = src[e];
        const int d = dst[e];
        const float* xrow = &x[(size_t)s * FEAT];
        float* grow = &agg[(size_t)d * FEAT];
        #pragma unroll
        for (int t = 0; t < 4; ++t) {
            const int n = 16 * t + ln;
            float v = acc[t][r] + bes[n] + xrow[n];
            v = fmaxf(v, 0.0f);
            atomicAdd(&grow[n], v);
        }
    }
}

// ---------------------------------------------------------------------------
// Node matmul: xout = maybe_relu((xin [+ agg]) @ W [+ bias])
// One wave per tile of 16 nodes; K=64 as 16 chunks of f32 WMMA 16x16x4.
// ---------------------------------------------------------------------------
template<bool USE_AGG, bool USE_BIAS, bool RELU>
__global__ __launch_bounds__(256)
void node_mm_kernel(const float* __restrict__ xin,   // [NN,64]
                    const float* __restrict__ agg,   // [NN,64] (optional)
                    const float* __restrict__ W,     // [64,64]
                    const float* __restrict__ bias,  // [64]  (optional)
                    float*       __restrict__ xout,  // [NN,64]
                    int nTiles)
{
    // Wp[kp][n][2] : Wp[kp*128 + 2n + j] = W[2*kp + j][n]
    __shared__ float Wp[FEAT * FEAT];                  // 16 KB
    for (int i = threadIdx.x; i < (FEAT / 2) * FEAT; i += 256) {
        const int kp = i >> 6;
        const int n  = i & 63;
        Wp[kp * 128 + 2 * n + 0] = W[(2 * kp    ) * FEAT + n];
        Wp[kp * 128 + 2 * n + 1] = W[(2 * kp + 1) * FEAT + n];
    }
    __syncthreads();

    int wave = (blockIdx.x * 256 + threadIdx.x) >> 5;
    int lane = threadIdx.x & 31;
    if (wave >= nTiles) return;

    const int nodeBase = wave * 16;
    const int hi = lane >> 4;
    const int ln = lane & 15;

    v8f acc0 = {0,0,0,0,0,0,0,0};
    v8f acc[4] = {acc0, acc0, acc0, acc0};

    const float* xrow = &xin[(size_t)(nodeBase + ln) * FEAT];
    const float* grow = USE_AGG ? &agg[(size_t)(nodeBase + ln) * FEAT] : nullptr;
    #pragma unroll
    for (int c = 0; c < 16; ++c) {
        const int kc = 4 * c + 2 * hi;                 // even
        v2f a = *(const v2f*)&xrow[kc];
        if (USE_AGG) {
            const v2f g = *(const v2f*)&grow[kc];
            a.x += g.x;
            a.y += g.y;
        }
        #pragma unroll
        for (int t = 0; t < 4; ++t) {
            const v2f b = *(const v2f*)&Wp[(kc >> 1) * 128 + 2 * (16 * t + ln)];
            acc[t] = __builtin_amdgcn_wmma_f32_16x16x4_f32(
                false, a, false, b, (short)0, acc[t], false, false);
        }
    }

    #pragma unroll
    for (int r = 0; r < 8; ++r) {
        const int m = r + 8 * hi;
        const size_t o = (size_t)(nodeBase + m) * FEAT;
        #pragma unroll
        for (int t = 0; t < 4; ++t) {
            const int n = 16 * t + ln;
            float v = acc[t][r];
            if (USE_BIAS) v += bias[n];
            if (RELU) v = fmaxf(v, 0.0f);
            xout[o + n] = v;
        }
    }
}

// ---------------------------------------------------------------------------
// GCN scatter: agg[dst] += dinv[src]*dinv[dst] * xw[src]
// One thread per (edge, 4-feature chunk): b128 load + 4 atomics.
// ---------------------------------------------------------------------------
__global__ void gcn_scatter_kernel(const float* __restrict__ xw,
                                   const int*   __restrict__ src,
                                   const int*   __restrict__ dst,
                                   const float* __restrict__ dinv,
                                   float*       __restrict__ agg,
                                   long total)                     // EE*16
{
    long i = (long)blockIdx.x * blockDim.x + threadIdx.x;
    long stride = (long)gridDim.x * blockDim.x;
    for (; i < total; i += stride) {
        const long e = i >> 4;
        const int  q = (int)(i & 15);                  // float4 chunk
        const int  s = src[e];
        const int  d = dst[e];
        const float w = dinv[s] * dinv[d];
        const float4 xv = ((const float4*)&xw[(size_t)s * FEAT])[q];
        float* grow = &agg[(size_t)d * FEAT + 4 * q];
        atomicAdd(&grow[0], w * xv.x);
        atomicAdd(&grow[1], w * xv.y);
        atomicAdd(&grow[2], w * xv.z);
        atomicAdd(&grow[3], w * xv.w);
    }
}

// h2 = relu(agg + dinv^2 * xw + bg)
__global__ void gcn_finalize_kernel(const float* __restrict__ agg,
                                    const float* __restrict__ xw,
                                    const float* __restrict__ dinv,
                                    const float* __restrict__ bg,
                                    float*       __restrict__ out,
                                    long total)                    // NN*64
{
    long i = (long)blockIdx.x * blockDim.x + threadIdx.x;
    if (i >= total) return;
    const long node = i >> 6;
    const int  f = (int)(i & 63);
    const float di = dinv[node];
    out[i] = fmaxf(agg[i] + di * di * xw[i] + bg[f], 0.0f);
}

// ---------------------------------------------------------------------------
// classifier: logits = h @ Wf + bf; out = log_softmax(logits)
// ---------------------------------------------------------------------------
__global__ __launch_bounds__(256)
void classify_kernel(const float* __restrict__ h,     // [NN,64]
                     const float* __restrict__ Wf,    // [64,7]
                     const float* __restrict__ bf,    // [7]
                     float*       __restrict__ out,   // [NN,7]
                     int n)
{
    __shared__ float Wfs[FEAT * NCLS];
    __shared__ float bfs[NCLS];
    for (int i = threadIdx.x; i < FEAT * NCLS; i += 256) Wfs[i] = Wf[i];
    if (threadIdx.x < NCLS) bfs[threadIdx.x] = bf[threadIdx.x];
    __syncthreads();

    int i = blockIdx.x * 256 + threadIdx.x;
    if (i >= n) return;

    float logits[NCLS];
    #pragma unroll
    for (int c = 0; c < NCLS; ++c) logits[c] = bfs[c];

    const float* hr = &h[(size_t)i * FEAT];
    #pragma unroll
    for (int k4 = 0; k4 < FEAT / 4; ++k4) {
        const float4 hv = ((const float4*)hr)[k4];
        const int k = 4 * k4;
        #pragma unroll
        for (int c = 0; c < NCLS; ++c) {
            float l = logits[c];
            l = fmaf(hv.x, Wfs[(k    ) * NCLS + c], l);
            l = fmaf(hv.y, Wfs[(k + 1) * NCLS + c], l);
            l = fmaf(hv.z, Wfs[(k + 2) * NCLS + c], l);
            l = fmaf(hv.w, Wfs[(k + 3) * NCLS + c], l);
            logits[c] = l;
        }
    }

    float mx = logits[0];
    #pragma unroll
    for (int c = 1; c < NCLS; ++c) mx = fmaxf(mx, logits[c]);
    float ssum = 0.0f;
    #pragma unroll
    for (int c = 0; c < NCLS; ++c) ssum += expf(logits[c] - mx);
    const float lse = logf(ssum);
    #pragma unroll
    for (int c = 0; c < NCLS; ++c) out[(size_t)i * NCLS + c] = logits[c] - mx - lse;
}

// ---------------------------------------------------------------------------
extern "C" void kernel_launch(void* const* d_in, const int* in_sizes, int n_in,
                              void* d_out, int out_size, void* d_ws, size_t ws_size,
                              hipStream_t stream)
{
    const float* x   = (const float*)d_in[0];
    const int*   ei  = (const int*)  d_in[1];   // [2,EE] int32
    const float* ea  = (const float*)d_in[2];
    const float* We0 = (const float*)d_in[3];
    const float* be0 = (const float*)d_in[4];
    const float* W0  = (const float*)d_in[5];
    const float* b0  = (const float*)d_in[6];
    const float* We1 = (const float*)d_in[7];
    const float* be1 = (const float*)d_in[8];
    const float* W1  = (const float*)d_in[9];
    const float* b1  = (const float*)d_in[10];
    const float* Wg  = (const float*)d_in[11];
    const float* bg  = (const float*)d_in[12];
    const float* Wf  = (const float*)d_in[13];
    const float* bf  = (const float*)d_in[14];
    float* out = (float*)d_out;

    const int* src = ei;
    const int* dst = ei + EE;

    // workspace layout
    float* agg  = (float*)d_ws;                 // [NN,64]
    float* bufA = agg  + (size_t)NN * FEAT;     // [NN,64]  h0 then xw
    float* bufB = bufA + (size_t)NN * FEAT;     // [NN,64]  h1 then h2
    float* dinv = bufB + (size_t)NN * FEAT;     // [NN]     deg then dinv

    const long nodeElems = (long)NN * FEAT;     // 6.4M
    const int  edgeTiles = EE / 16;             // 100000
    const int  nodeTiles = (NN + 15) / 16;      // 6250
    const int  edgeBlocks = (edgeTiles * 32 + 255) / 256;
    const int  nodeBlocks = (nodeTiles * 32 + 255) / 256;
    const int  zeroBlocks = (int)((nodeElems + 255) / 256);

    // ---- GINE layer 0 --------------------------------------------------
    zero_kernel<<<zeroBlocks, 256, 0, stream>>>(agg, nodeElems);
    gine_edge_kernel<<<edgeBlocks, 256, 0, stream>>>(x, ea, We0, be0, src, dst, agg, edgeTiles);
    node_mm_kernel<true, true, true><<<nodeBlocks, 256, 0, stream>>>(x, agg, W0, b0, bufA, nodeTiles);

    // ---- GINE layer 1 --------------------------------------------------
    zero_kernel<<<zeroBlocks, 256, 0, stream>>>(agg, nodeElems);
    gine_edge_kernel<<<edgeBlocks, 256, 0, stream>>>(bufA, ea, We1, be1, src, dst, agg, edgeTiles);
    node_mm_kernel<true, true, true><<<nodeBlocks, 256, 0, stream>>>(bufA, agg, W1, b1, bufB, nodeTiles);

    // ---- GCN layer -----------------------------------------------------
    // xw = h1 @ Wg  (into bufA; h0 no longer needed)
    node_mm_kernel<false, false, false><<<nodeBlocks, 256, 0, stream>>>(bufB, nullptr, Wg, nullptr, bufA, nodeTiles);

    zero_kernel<<<(NN + 255) / 256, 256, 0, stream>>>(dinv, NN);
    deg_kernel<<<4096, 256, 0, stream>>>(dst, dinv, EE);
    dinv_kernel<<<(NN + 255) / 256, 256, 0, stream>>>(dinv, NN);

    zero_kernel<<<zeroBlocks, 256, 0, stream>>>(agg, nodeElems);
    const long scatterTotal = (long)EE * (FEAT / 4);
    gcn_scatter_kernel<<<65536, 256, 0, stream>>>(bufA, src, dst, dinv, agg, scatterTotal);
    gcn_finalize_kernel<<<(int)((nodeElems + 255) / 256), 256, 0, stream>>>(agg, bufA, dinv, bg, bufB, nodeElems);

    // ---- classifier + log_softmax -------------------------------------
    classify_kernel<<<(NN + 255) / 256, 256, 0, stream>>>(bufB, Wf, bf, out, NN);
}